// SpectralConv3d_76192719831980
// MI455X (gfx1250) — compile-verified
//
#include <hip/hip_runtime.h>
#include <math.h>

typedef __attribute__((ext_vector_type(2))) float v2f;
typedef __attribute__((ext_vector_type(8))) float v8f;

#define WMMA4(a, b, c) \
  __builtin_amdgcn_wmma_f32_16x16x4_f32(false, (a), false, (b), (short)0, (c), false, false)

#if defined(__AMDGCN__) && __has_builtin(__builtin_amdgcn_global_load_async_to_lds_b128)
#define ASYNC_LDS 1
#define AS1 __attribute__((address_space(1)))
#define AS3 __attribute__((address_space(3)))
typedef int v4i __attribute__((vector_size(4 * sizeof(int))));
__device__ __forceinline__ void async_wait0() {
#if __has_builtin(__builtin_amdgcn_s_wait_asynccnt)
  __builtin_amdgcn_s_wait_asynccnt(0);
#else
  asm volatile("s_wait_asynccnt 0" ::: "memory");
#endif
}
#endif

// Problem sizes (fixed by reference): B=4, Cin=Cout=32, D=64, modes=16
// BC = 128 combined batch*channel rows.
// Workspace float offsets:
//   mats  @ 0      : Cf[64][16] @0, Eip[16][48] @1024, Ezr[16][17] @2048, Ezi @2560
//   A     @ 4096   : max 8,388,608 floats (t1 / t3 / t5)
//   Bf    @ A+8388608 : max 2,230,272 floats (t2 / t4 / t6)

__global__ void init_mats(float* mats) {
  const double PI = 3.14159265358979323846;
  int t = threadIdx.x;
  if (t < 1024) {  // Cf[j][k], j=0..63, k=0..15  (hfft of real input, N=126)
    int j = t >> 4, k = t & 15;
    double w = (j == 0 || j == 63) ? 1.0 : 2.0;
    mats[t] = (float)(w * cos(2.0 * PI * (double)j * (double)k / 126.0));
  }
  if (t < 768) {   // Eip[k1][u] padded to 48 cols, u valid 0..32 (ihfft n=64)
    int k1 = t / 48, u = t % 48;
    mats[1024 + t] = (u < 33)
        ? (float)(cos(2.0 * PI * (double)k1 * (double)u / 64.0) / 64.0) : 0.0f;
  }
  if (t < 272) {   // Ez real/imag [k3][u3], u3=0..16 (ihfft n=33, conj(rfft)/n)
    int k3 = t / 17, u = t % 17;
    mats[2048 + t] = (float)(cos(2.0 * PI * (double)k3 * (double)u / 33.0) / 33.0);
    mats[2560 + t] = (float)(sin(2.0 * PI * (double)k3 * (double)u / 33.0) / 33.0);
  }
}

// Stage a contiguous 16x64-float tile (4KB) into this wave's LDS region.
// Prefer CDNA5 async global->LDS DMA (ASYNCcnt) over register staging.
__device__ __forceinline__ void stage_tile(const float* __restrict__ gsrc,
                                           float* __restrict__ ldst, int lane) {
#ifdef ASYNC_LDS
  AS3 v4i* l = (AS3 v4i*)(ldst + lane * 4);
  AS1 v4i* g = (AS1 v4i*)(void*)(gsrc + lane * 4);
#pragma unroll
  for (int i = 0; i < 8; ++i)
    __builtin_amdgcn_global_load_async_to_lds_b128(g + i * 32, l + i * 32, 0, 0);
  async_wait0();
#else
  const float4* g4 = (const float4*)gsrc;
  float4* l4 = (float4*)ldst;
#pragma unroll
  for (int i = 0; i < 8; ++i) l4[lane + 32 * i] = g4[lane + 32 * i];
#endif
}

// S1: contract z: t1[row][k3] = sum_j3 x[row][j3]*Cf[j3][k3], row = (bc,j1,j2)
__global__ void __launch_bounds__(256) dct_z(const float* __restrict__ x,
                                             const float* __restrict__ mats,
                                             float* __restrict__ t1) {
  __shared__ float ldsCf[1024];
  __shared__ float tile[8][1024];
  int tid = threadIdx.x, lane = tid & 31, wid = tid >> 5;
  for (int i = tid; i < 1024; i += 256) ldsCf[i] = mats[i];
  int wave = blockIdx.x * 8 + wid;                        // 0..32767
  stage_tile(x + (size_t)wave * 1024, tile[wid], lane);   // 16x64 contiguous
  __syncthreads();
  const float* lt = tile[wid];
  int r = lane & 15, koff = (lane >> 4) << 1;
  v8f acc = {};
#pragma unroll
  for (int k0 = 0; k0 < 64; k0 += 4) {
    v2f a = *(const v2f*)&lt[r * 64 + k0 + koff];          // A[m=r][k,k+1]
    v2f b;
    b.x = ldsCf[(k0 + koff) * 16 + r];                     // B[k][n=r]
    b.y = ldsCf[(k0 + koff + 1) * 16 + r];
    acc = WMMA4(a, b, acc);
  }
  float* dst = t1 + (size_t)wave * 256;
  int mh = (lane >> 4) * 8, n = lane & 15;
#pragma unroll
  for (int v = 0; v < 8; ++v) dst[(mh + v) * 16 + n] = acc[v];
}

// S2: contract y: t2[bc,j1][k2][k3] = sum_j2 Cf[j2][k2]*t1[bc,j1,j2,k3]
__global__ void __launch_bounds__(256) dct_y(const float* __restrict__ t1,
                                             const float* __restrict__ mats,
                                             float* __restrict__ t2) {
  __shared__ float ldsCf[1024];
  __shared__ float tile[8][1024];
  int tid = threadIdx.x, lane = tid & 31, wid = tid >> 5;
  for (int i = tid; i < 1024; i += 256) ldsCf[i] = mats[i];
  int wave = blockIdx.x * 8 + wid;                        // bc*64 + j1
  stage_tile(t1 + (size_t)wave * 1024, tile[wid], lane);  // [j2][k3] 64x16
  __syncthreads();
  const float* lt = tile[wid];
  int r = lane & 15, koff = (lane >> 4) << 1;
  v8f acc = {};
#pragma unroll
  for (int k0 = 0; k0 < 64; k0 += 4) {
    v2f a, b;
    a.x = ldsCf[(k0 + koff) * 16 + r];                    // A[m=k2][k=j2] = Cf[j2][k2]
    a.y = ldsCf[(k0 + koff + 1) * 16 + r];
    b.x = lt[(k0 + koff) * 16 + r];                       // B[j2][n=k3]
    b.y = lt[(k0 + koff + 1) * 16 + r];
    acc = WMMA4(a, b, acc);
  }
  float* dst = t2 + (size_t)wave * 256;
  int mh = (lane >> 4) * 8, n = lane & 15;
#pragma unroll
  for (int v = 0; v < 8; ++v) dst[(mh + v) * 16 + n] = acc[v];
}

// S3: contract x: t3[bc][k1][k2][k3] = sum_j1 Cf[j1][k1]*t2[bc][j1][k2][k3]
__global__ void __launch_bounds__(256) dct_x(const float* __restrict__ t2,
                                             const float* __restrict__ mats,
                                             float* __restrict__ t3) {
  __shared__ float ldsCf[1024];
  int tid = threadIdx.x, lane = tid & 31, wid = tid >> 5;
  for (int i = tid; i < 1024; i += 256) ldsCf[i] = mats[i];
  __syncthreads();
  int wave = blockIdx.x * 8 + wid;                        // bc*16 + k2, 0..2047
  int bc = wave >> 4, k2 = wave & 15;
  const float* src = t2 + (size_t)bc * 16384;             // j1 stride 256
  int r = lane & 15, koff = (lane >> 4) << 1;
  v8f acc = {};
#pragma unroll
  for (int k0 = 0; k0 < 64; k0 += 4) {
    v2f a, b;
    a.x = ldsCf[(k0 + koff) * 16 + r];
    a.y = ldsCf[(k0 + koff + 1) * 16 + r];
    b.x = src[(size_t)(k0 + koff) * 256 + k2 * 16 + r];
    b.y = src[(size_t)(k0 + koff + 1) * 256 + k2 * 16 + r];
    acc = WMMA4(a, b, acc);
  }
  float* dst = t3 + (size_t)bc * 4096 + k2 * 16;          // [k1][k2][k3]
  int mh = (lane >> 4) * 8, n = lane & 15;
#pragma unroll
  for (int v = 0; v < 8; ++v) dst[(size_t)(mh + v) * 256 + n] = acc[v];
}

// S4: channel mixing (M=4 -> VALU): t4[b][o][m] = sum_i t3[b*32+i][m]*w[i][o][m]
__global__ void __launch_bounds__(256) mix_ch(const float* __restrict__ t3,
                                              const float* __restrict__ w,
                                              float* __restrict__ t4) {
  int idx = blockIdx.x * 256 + threadIdx.x;               // 524288
  int m = idx & 4095;
  int o = (idx >> 12) & 31;
  int b = idx >> 17;
  float acc = 0.0f;
#pragma unroll 4
  for (int i = 0; i < 32; ++i)
    acc = fmaf(t3[(size_t)(b * 32 + i) * 4096 + m],
               w[(size_t)(i * 32 + o) * 4096 + m], acc);
  t4[idx] = acc;
}

// S5: inverse along k1 -> u1 (0..32): t5[bo][u1][k2][k3]
__global__ void __launch_bounds__(256) inv_x(const float* __restrict__ t4,
                                             const float* __restrict__ mats,
                                             float* __restrict__ t5) {
  __shared__ float ldsE[768];
  int tid = threadIdx.x, lane = tid & 31, wid = tid >> 5;
  for (int i = tid; i < 768; i += 256) ldsE[i] = mats[1024 + i];
  __syncthreads();
  int wave = blockIdx.x * 8 + wid;                        // (bo*16+k2)*3+ut, 6144
  int ut = wave % 3;
  int bk = wave / 3;
  int bo = bk >> 4, k2 = bk & 15;
  const float* src = t4 + (size_t)bo * 4096;              // [k1][k2][k3]
  int r = lane & 15, koff = (lane >> 4) << 1, ubase = ut * 16;
  v8f acc = {};
#pragma unroll
  for (int k0 = 0; k0 < 16; k0 += 4) {
    v2f a, b;
    a.x = ldsE[(k0 + koff) * 48 + ubase + r];             // A[m=u1][k=k1]=Eip[k1][u1]
    a.y = ldsE[(k0 + koff + 1) * 48 + ubase + r];
    b.x = src[(k0 + koff) * 256 + k2 * 16 + r];
    b.y = src[(k0 + koff + 1) * 256 + k2 * 16 + r];
    acc = WMMA4(a, b, acc);
  }
  int mh = (lane >> 4) * 8, n = lane & 15;
#pragma unroll
  for (int v = 0; v < 8; ++v) {
    int u1 = ubase + mh + v;
    if (u1 < 33) t5[(((size_t)bo * 33 + u1) * 16 + k2) * 16 + n] = acc[v];
  }
}

// S6: inverse along k2 -> u2 (0..32): t6[bo][u1][u2][k3]
__global__ void __launch_bounds__(256) inv_y(const float* __restrict__ t5,
                                             const float* __restrict__ mats,
                                             float* __restrict__ t6) {
  __shared__ float ldsE[768];
  int tid = threadIdx.x, lane = tid & 31, wid = tid >> 5;
  for (int i = tid; i < 768; i += 256) ldsE[i] = mats[1024 + i];
  __syncthreads();
  int wave = blockIdx.x * 8 + wid;                        // (bo*33+u1)*3+ut, 12672
  int ut = wave % 3;
  int bu = wave / 3;                                      // bo*33+u1, 0..4223
  const float* src = t5 + (size_t)bu * 256;               // [k2][k3] contiguous
  int r = lane & 15, koff = (lane >> 4) << 1, ubase = ut * 16;
  v8f acc = {};
#pragma unroll
  for (int k0 = 0; k0 < 16; k0 += 4) {
    v2f a, b;
    a.x = ldsE[(k0 + koff) * 48 + ubase + r];
    a.y = ldsE[(k0 + koff + 1) * 48 + ubase + r];
    b.x = src[(k0 + koff) * 16 + r];
    b.y = src[(k0 + koff + 1) * 16 + r];
    acc = WMMA4(a, b, acc);
  }
  int mh = (lane >> 4) * 8, n = lane & 15;
#pragma unroll
  for (int v = 0; v < 8; ++v) {
    int u2 = ubase + mh + v;
    if (u2 < 33) t6[((size_t)bu * 33 + u2) * 16 + n] = acc[v];
  }
}

// S7: inverse along k3 -> u3 (complex, 0..16), write [..][u3][re,im]
__global__ void __launch_bounds__(256) inv_z(const float* __restrict__ t6,
                                             const float* __restrict__ mats,
                                             float* __restrict__ out) {
  size_t idx = (size_t)blockIdx.x * 256 + threadIdx.x;
  const size_t TOT = (size_t)128 * 33 * 33 * 17;
  if (idx >= TOT) return;
  int u3 = (int)(idx % 17);
  size_t row = idx / 17;                                  // (bo*33+u1)*33+u2
  const float* src = t6 + row * 16;
  const float* Ezr = mats + 2048;
  const float* Ezi = mats + 2560;
  float re = 0.0f, im = 0.0f;
#pragma unroll
  for (int k = 0; k < 16; ++k) {
    float v = src[k];
    re = fmaf(v, Ezr[k * 17 + u3], re);
    im = fmaf(v, Ezi[k * 17 + u3], im);
  }
  out[idx * 2 + 0] = re;
  out[idx * 2 + 1] = im;
}

extern "C" void kernel_launch(void* const* d_in, const int* in_sizes, int n_in,
                              void* d_out, int out_size, void* d_ws, size_t ws_size,
                              hipStream_t stream) {
  const float* x = (const float*)d_in[0];     // (4,32,64,64,64)
  const float* w = (const float*)d_in[1];     // (32,32,16,16,16)
  float* out = (float*)d_out;                 // (4,32,33,33,17,2)
  float* ws = (float*)d_ws;
  float* mats = ws;
  float* A = ws + 4096;                       // 8,388,608 floats
  float* Bf = A + 8388608;                    // 2,230,272 floats

  init_mats<<<1, 1024, 0, stream>>>(mats);
  dct_z<<<4096, 256, 0, stream>>>(x, mats, A);      // t1 in A
  dct_y<<<1024, 256, 0, stream>>>(A, mats, Bf);     // t2 in Bf
  dct_x<<<256, 256, 0, stream>>>(Bf, mats, A);      // t3 in A
  mix_ch<<<2048, 256, 0, stream>>>(A, w, Bf);       // t4 in Bf
  inv_x<<<768, 256, 0, stream>>>(Bf, mats, A);      // t5 in A
  inv_y<<<1584, 256, 0, stream>>>(A, mats, Bf);     // t6 in Bf
  inv_z<<<9257, 256, 0, stream>>>(Bf, mats, out);
}